// MinibatchDiscrimination_33414845562989
// MI455X (gfx1250) — compile-verified
//
#include <hip/hip_runtime.h>

typedef __attribute__((ext_vector_type(2))) float v2f;
typedef __attribute__((ext_vector_type(8))) float v8f;

#define NROWS 512      // N
#define KDIM  1024     // IN_FEATURES
#define NCOLS 1024     // OUT_FEATURES * KERNEL_DIMS
#define OUTF  64
#define KD    16
#define OUT_STRIDE 1088  // IN_FEATURES + OUT_FEATURES

// ---------------------------------------------------------------------------
// Kernel 1: M = X @ Tr  (fp32 WMMA 16x16x4), X:[512,1024], Tr:[1024,1024]
// One wave per 16x16 tile of M. 2048 tiles = 256 blocks x 8 waves.
// ---------------------------------------------------------------------------
__global__ __launch_bounds__(256) void mbd_gemm_wmma(
    const float* __restrict__ X, const float* __restrict__ T,
    float* __restrict__ M)
{
    const int lane = threadIdx.x & 31;
    const int wave = threadIdx.x >> 5;
    const int w    = blockIdx.x * 8 + wave;   // 0..2047
    const int mt   = w >> 6;                  // 0..31  (row tile)
    const int nt   = w & 63;                  // 0..63  (col tile)
    const int half = lane >> 4;               // 0 or 1
    const int ln16 = lane & 15;
    const int kh   = half * 2;                // K sub-offset for A/B frags

    const float* Arow  = X + (mt * 16 + ln16) * KDIM; // this lane's A row
    const float* Bbase = T + nt * 16 + ln16;          // this lane's B column

    v8f c = {};
    #pragma unroll 8
    for (int k = 0; k < KDIM; k += 4) {
        // A 16x4: VGPR0 = K(kh), VGPR1 = K(kh+1); rows striped over lanes 0-15
        v2f a = *(const v2f*)(Arow + k + kh);
        // B 4x16: VGPR0 = row K(kh), VGPR1 = row K(kh+1); cols striped over lanes
        v2f b;
        b.x = Bbase[(k + kh)     * NCOLS];
        b.y = Bbase[(k + kh + 1) * NCOLS];
        c = __builtin_amdgcn_wmma_f32_16x16x4_f32(
                /*neg_a=*/false, a, /*neg_b=*/false, b,
                /*c_mod=*/(short)0, c, /*reuse_a=*/false, /*reuse_b=*/false);
    }

    // C/D layout: VGPR r -> row r (lanes 0-15) / row r+8 (lanes 16-31)
    const int row0 = mt * 16 + half * 8;
    const int col  = nt * 16 + ln16;
    #pragma unroll
    for (int r = 0; r < 8; ++r)
        M[(row0 + r) * NCOLS + col] = c[r];
}

// ---------------------------------------------------------------------------
// Kernel 2: o_b[i,o] = sum_{j != i} exp(-sum_k |M[i,o,k]-M[j,o,k]|)
// Grid (NROWS/256, OUTF); all 512 rows of feature o staged in 32KB LDS.
// Diagonal term is exp(0)=1 -> subtract 1 instead of masking.
// ---------------------------------------------------------------------------
__global__ __launch_bounds__(256) void mbd_pairwise(
    const float* __restrict__ M, float* __restrict__ out)
{
    __shared__ float sj[NROWS * KD];          // 32 KB

    const int o = blockIdx.y;
    const int i = blockIdx.x * 256 + threadIdx.x;

    // Stage all rows j for feature o: 512 rows x 16 floats, float4 copies
    for (int j = threadIdx.x; j < NROWS; j += 256) {
        const float4* src = (const float4*)(M + j * NCOLS + o * KD);
        float4*       dst = (float4*)(sj + j * KD);
        dst[0] = src[0]; dst[1] = src[1]; dst[2] = src[2]; dst[3] = src[3];
    }
    __syncthreads();

    float mi[KD];
    #pragma unroll
    for (int k = 0; k < KD; ++k) mi[k] = sj[i * KD + k];

    float acc = 0.0f;
    for (int j = 0; j < NROWS; ++j) {
        float norm = 0.0f;
        #pragma unroll
        for (int k = 0; k < KD; ++k)
            norm += fabsf(mi[k] - sj[j * KD + k]);   // broadcast LDS reads
        acc += __expf(-norm);
    }
    acc -= 1.0f;                                     // remove j == i

    out[i * OUT_STRIDE + KDIM + o] = acc;
}

// ---------------------------------------------------------------------------
// Kernel 3: passthrough copy of x into out[:, 0:1024] (row stride 1088)
// ---------------------------------------------------------------------------
__global__ __launch_bounds__(256) void mbd_copy_x(
    const float* __restrict__ X, float* __restrict__ out)
{
    const int t   = blockIdx.x * 256 + threadIdx.x;  // 512*256 float4 elements
    const int row = t >> 8;
    const int c4  = t & 255;
    ((float4*)(out + row * OUT_STRIDE))[c4] =
        ((const float4*)(X + row * KDIM))[c4];
}

// ---------------------------------------------------------------------------
extern "C" void kernel_launch(void* const* d_in, const int* in_sizes, int n_in,
                              void* d_out, int out_size, void* d_ws, size_t ws_size,
                              hipStream_t stream)
{
    const float* x = (const float*)d_in[0];   // [512, 1024]
    const float* T = (const float*)d_in[1];   // [1024, 64, 16] == [1024, 1024]
    float* out = (float*)d_out;               // [512, 1088]
    float* M   = (float*)d_ws;                // [512, 1024] scratch (2 MB)

    mbd_gemm_wmma<<<256, 256, 0, stream>>>(x, T, M);
    mbd_copy_x  <<<512, 256, 0, stream>>>(x, out);
    mbd_pairwise<<<dim3(2, 64), 256, 0, stream>>>(M, out);
}